// GPT4RecommendationBaseModel_54039278519093
// MI455X (gfx1250) — compile-verified
//
#include <hip/hip_runtime.h>
#include <stdint.h>

// ---- problem constants (from reference) ----
#define VOCAB   50257
#define NUSERS  10000
#define NITEMS  50000
#define DIM     768          // floats per row (3072 bytes)
#define KC0     512
#define KC1     512
#define BATCH   16
#define SEQ     2048

typedef float v4f __attribute__((ext_vector_type(4)));

// ============================================================================
// Kernel A: per-batch cluster presence + segmented scan + inverse row map.
// One 1024-thread block per batch. rowmap[b*1024 + r] =
//   -1            -> zero row
//   k  in [0,512) -> centers0[k]
//   k  in [512,1024) -> centers1[k-512]
// ============================================================================
__global__ __launch_bounds__(1024)
void build_rowmap_kernel(const int* __restrict__ ids,
                         const int* __restrict__ labels0,
                         const int* __restrict__ labels1,
                         int* __restrict__ rowmap) {
  __shared__ int pres[KC0 + KC1];
  __shared__ int scan[KC0 + KC1];
  const int b   = blockIdx.x;
  const int tid = threadIdx.x;            // 0..1023

  pres[tid] = 0;
  __syncthreads();

  // presence bitmap: tokens >= VOCAB mark their cluster labels
  for (int s = tid; s < SEQ; s += 1024) {
    int ui = ids[b * SEQ + s] - VOCAB;
    if (ui >= 0) {                        // user or item token, ui in [0, 60000)
      pres[labels0[ui]]       = 1;        // racing writes of 1 are benign
      pres[KC0 + labels1[ui]] = 1;
    }
  }
  __syncthreads();

  // segmented (512-wide) inclusive scan over both tables at once
  int p = pres[tid];
  scan[tid] = p;
  __syncthreads();
  for (int off = 1; off < 512; off <<= 1) {
    int add = ((tid & 511) >= off) ? scan[tid - off] : 0;
    __syncthreads();
    scan[tid] += add;
    __syncthreads();
  }

  const int sum0 = scan[KC0 - 1];
  const int c0   = (sum0 > 0) ? sum0 : 1;   // reference: cnt = max(sum, 1)
  const int pos  = scan[tid] - p;           // exclusive position within segment
  const int row  = (tid < KC0) ? pos : (c0 + pos);

  // invert the packing: reuse pres[] as row->source map (LDS-ordered)
  __syncthreads();
  pres[tid] = -1;
  __syncthreads();
  if (p) pres[row] = tid;                   // rows are unique (disjoint ranges)
  __syncthreads();
  rowmap[b * (KC0 + KC1) + tid] = pres[tid];
}

// ============================================================================
// Kernel B: materialize a_node rows. One block (192 threads = 6 waves) per
// output row. Gathered rows go through LDS via CDNA5 async DMA:
//   global_load_async_to_lds_b128  ->  s_wait_asynccnt  ->
//   global_store_async_from_lds_b128
// Each wave moves its own 512B chunk, so no cross-wave barrier is needed.
// ============================================================================
__global__ __launch_bounds__(192)
void materialize_rows_kernel(const int* __restrict__ rowmap,
                             const float* __restrict__ cen0,
                             const float* __restrict__ cen1,
                             float* __restrict__ out_a) {
  __shared__ __attribute__((aligned(16))) float buf[DIM];
  const int r   = blockIdx.x;                 // 0 .. B*1024-1
  const int tid = threadIdx.x;                // 0..191, 16B per lane
  float* dst = out_a + (size_t)r * DIM;

  const int v = rowmap[r];                    // uniform per block
  if (v < 0) {
    v4f z = (v4f)0.0f;
    __builtin_nontemporal_store(z, (v4f*)(dst + tid * 4));
    return;
  }
  const float* src = (v < KC0) ? (cen0 + (size_t)v * DIM)
                               : (cen1 + (size_t)(v - KC0) * DIM);

  uint64_t gsrc = (uint64_t)(uintptr_t)(src + tid * 4);
  uint64_t gdst = (uint64_t)(uintptr_t)(dst + tid * 4);
  uint32_t lds  = (uint32_t)(uintptr_t)buf + (uint32_t)(tid * 16);

  // async DMA global -> LDS (16B per lane, 512B per wave-instruction)
  asm volatile("global_load_async_to_lds_b128 %0, %1, off"
               :: "v"(lds), "v"(gsrc) : "memory");
  asm volatile("s_wait_asynccnt 0" ::: "memory");
  // async DMA LDS -> global
  asm volatile("global_store_async_from_lds_b128 %0, %1, off"
               :: "v"(gdst), "v"(lds) : "memory");
  asm volatile("s_wait_asynccnt 0" ::: "memory");
}

// ============================================================================
// Kernel C: masked 3-table embedding gather + node_list.
// 64 lanes per token row (3 x b128 per lane), 4 tokens per 256-thread block.
// Non-temporal stores keep the embedding tables resident in L2.
// ============================================================================
__global__ __launch_bounds__(256)
void embed_gather_kernel(const int* __restrict__ ids,
                         const float* __restrict__ wte,
                         const float* __restrict__ uemb,
                         const float* __restrict__ iemb,
                         float* __restrict__ out_emb,
                         float* __restrict__ out_node) {
  const int g    = blockIdx.x * 4 + (threadIdx.x >> 6);  // token in [0, B*S)
  const int lane = threadIdx.x & 63;

  const int id = ids[g];
  const float* src;
  float nl;
  if (id < VOCAB) {
    src = wte + (size_t)id * DIM;                  nl = 0.0f;
  } else if (id < VOCAB + NUSERS) {
    src = uemb + (size_t)(id - VOCAB) * DIM;       nl = 1.0f;
  } else {
    src = iemb + (size_t)(id - VOCAB - NUSERS) * DIM; nl = 1.0f;
  }
  __builtin_prefetch(src, 0, 0);                   // global_prefetch_b8

  if (lane == 0) out_node[g] = nl;                 // node_list (float buffer)

  float* dst = out_emb + (size_t)g * DIM;
#pragma unroll
  for (int i = 0; i < 3; ++i) {
    const int off = (lane + 64 * i) * 4;           // 768 floats = 64*3 x float4
    v4f val = *(const v4f*)(src + off);
    __builtin_nontemporal_store(val, (v4f*)(dst + off));
  }
}

// ============================================================================
extern "C" void kernel_launch(void* const* d_in, const int* in_sizes, int n_in,
                              void* d_out, int out_size, void* d_ws, size_t ws_size,
                              hipStream_t stream) {
  const int*   ids  = (const int*)  d_in[0];   // [B,S]
  const float* wte  = (const float*)d_in[1];   // [VOCAB, D]
  const float* uemb = (const float*)d_in[2];   // [NUSERS, D]
  const float* iemb = (const float*)d_in[3];   // [NITEMS, D]
  const float* cen0 = (const float*)d_in[4];   // [K0, D]
  const float* cen1 = (const float*)d_in[5];   // [K1, D]
  const int*   lab0 = (const int*)  d_in[6];   // [NUSERS+NITEMS]
  const int*   lab1 = (const int*)  d_in[7];   // [NUSERS+NITEMS]

  // outputs concatenated flat in return order, all float32:
  float* out_emb  = (float*)d_out;                             // B*S*D
  float* out_node = out_emb + (size_t)BATCH * SEQ * DIM;       // B*S
  float* out_a    = out_node + (size_t)BATCH * SEQ;            // B*1024*D

  int* rowmap = (int*)d_ws;                                    // B*1024 ints (64 KB)

  build_rowmap_kernel<<<BATCH, 1024, 0, stream>>>(ids, lab0, lab1, rowmap);
  materialize_rows_kernel<<<BATCH * (KC0 + KC1), 192, 0, stream>>>(
      rowmap, cen0, cen1, out_a);
  embed_gather_kernel<<<(BATCH * SEQ) / 4, 256, 0, stream>>>(
      ids, wte, uemb, iemb, out_emb, out_node);
}